// Minimax_Conv2D_57174604644628
// MI455X (gfx1250) — compile-verified
//
#include <hip/hip_runtime.h>
#include <stdint.h>

// Problem constants (match reference)
#define CC 64    // in channels
#define OO 128   // out channels
#define HH 64
#define WW 64
#define BB 16

#if defined(__gfx1250__) && __has_builtin(__builtin_amdgcn_global_load_async_to_lds_b128)
#define USE_ASYNC_LDS 1
#else
#define USE_ASYNC_LDS 0
#endif

#if USE_ASYNC_LDS
typedef int v4i_cdna5 __attribute__((ext_vector_type(4)));
typedef __attribute__((address_space(1))) v4i_cdna5* gas_v4i;
typedef __attribute__((address_space(3))) v4i_cdna5* las_v4i;
#endif

__device__ __forceinline__ void stage16(const float* __restrict__ g, float* l) {
#if USE_ASYNC_LDS
  // async DMA: 16 bytes per lane, global -> LDS, tracked by ASYNCcnt
  __builtin_amdgcn_global_load_async_to_lds_b128(
      (gas_v4i)(uintptr_t)g,
      (las_v4i)(uint32_t)(uintptr_t)l,
      /*offset=*/0, /*cpol=*/0);
#else
  *reinterpret_cast<float4*>(l) = *reinterpret_cast<const float4*>(g);
#endif
}

__device__ __forceinline__ void wait_stage() {
#if USE_ASYNC_LDS
#if __has_builtin(__builtin_amdgcn_s_wait_asynccnt)
  __builtin_amdgcn_s_wait_asynccnt(0);
#else
  asm volatile("s_wait_asynccnt 0" ::: "memory");
#endif
#endif
}

// One block: batch b = blockIdx.z, output row h0 = blockIdx.x,
// output channels [blockIdx.y*64, +64). Stages 3 clamped input rows for all
// 64 input channels into LDS, then computes the 9-gather minimax per pixel.
__global__ __launch_bounds__(256) void minimax_conv2d_kernel(
    const float* __restrict__ x,     // [B, C, H, W]
    const float* __restrict__ w1,    // [O, 9]
    const float* __restrict__ w2,    // [O, 3]
    const int*   __restrict__ conn,  // [O*9], values in [0, C*9)
    float* __restrict__ out)         // [B, O, H, W]
{
  // LDS tile: [c][r][w], r = 0..2 are source rows clamp(h0-1+r)
  __shared__ __align__(16) float smem[CC * 3 * WW];   // 48 KB

  const int tid   = threadIdx.x;
  const int h0    = blockIdx.x;
  const int obase = blockIdx.y << 6;  // 0 or 64
  const int b     = blockIdx.z;

  // ---- stage input tile: 64ch x 3rows x 64cols = 3072 float4 transfers ----
  const float* xb = x + (size_t)b * (CC * HH * WW);
  #pragma unroll
  for (int it = 0; it < (CC * 3 * WW) / 4 / 256; ++it) {  // 12 iters
    int q   = tid + (it << 8);        // float4 index in [0, 3072)
    int w4  = q & 15;                 // 16 float4 per row
    int r   = (q >> 4) % 3;
    int c   = q / 48;
    int srow = h0 - 1 + r;            // replicate padding on H
    srow = srow < 0 ? 0 : (srow > HH - 1 ? HH - 1 : srow);
    const float* src = xb + ((size_t)(c * HH + srow)) * WW + (w4 << 2);
    float* dst = smem + (q << 2);     // 4*q floats == (c*3+r)*64 + 4*w4
    stage16(src, dst);
  }
  wait_stage();
  __syncthreads();

  const int wave = tid >> 5;
  const int lane = tid & 31;

  #pragma unroll 1
  for (int oi = 0; oi < 8; ++oi) {
    // wave-uniform output channel (force scalar so conn/w1/w2 use s_loads)
    const int o = obase + __builtin_amdgcn_readfirstlane(wave + (oi << 3));

    // decode the 9 connections once per channel (all wave-uniform)
    int   rowb[9];
    int   dws[9];
    float w1v[9];
    #pragma unroll
    for (int j = 0; j < 9; ++j) {
      int idx = conn[o * 9 + j];      // = c*9 + kh*3 + kw
      int c   = idx / 9;
      int kp  = idx - c * 9;
      int kh  = kp / 3;
      int kw  = kp - kh * 3;
      rowb[j] = (c * 3 + kh) * WW;    // LDS row base
      dws[j]  = kw - 1;               // column shift, pad_l = 1
      w1v[j]  = w1[o * 9 + j];
    }
    const float w2v0 = w2[o * 3 + 0];
    const float w2v1 = w2[o * 3 + 1];
    const float w2v2 = w2[o * 3 + 2];

    const size_t ob = (((size_t)b * OO + o) * HH + h0) * WW;

    #pragma unroll
    for (int half = 0; half < 2; ++half) {
      const int w = (half << 5) + lane;
      float v[9];
      #pragma unroll
      for (int j = 0; j < 9; ++j) {
        int ws = w + dws[j];                              // replicate pad on W
        ws = ws < 0 ? 0 : (ws > WW - 1 ? WW - 1 : ws);    // -> v_med3
        v[j] = smem[rowb[j] + ws] - w1v[j];
      }
      float m0 = fmaxf(fmaxf(v[0], v[1]), v[2]) - w2v0;
      float m1 = fmaxf(fmaxf(v[3], v[4]), v[5]) - w2v1;
      float m2 = fmaxf(fmaxf(v[6], v[7]), v[8]) - w2v2;
      out[ob + w] = fminf(fminf(m0, m1), m2);
    }
  }
}

extern "C" void kernel_launch(void* const* d_in, const int* in_sizes, int n_in,
                              void* d_out, int out_size, void* d_ws, size_t ws_size,
                              hipStream_t stream) {
  const float* x    = (const float*)d_in[0];  // [16,64,64,64] f32
  const float* w1   = (const float*)d_in[1];  // [128,9] f32
  const float* w2   = (const float*)d_in[2];  // [128,3] f32
  const int*   conn = (const int*)  d_in[3];  // [1152] i32
  float* out = (float*)d_out;                 // [16,128,64,64] f32

  dim3 grid(HH, OO / 64, BB);                 // (64, 2, 16)
  dim3 block(256, 1, 1);
  minimax_conv2d_kernel<<<grid, block, 0, stream>>>(x, w1, w2, conn, out);
}